// FreqProj_80049600463644
// MI455X (gfx1250) — compile-verified
//
#include <hip/hip_runtime.h>
#include <stdint.h>

// ---------------- problem constants (from reference) ----------------
#define FGRID 1024
#define F2    (FGRID * FGRID)     // 1,048,576 frequencies
#define BATCH 32
#define CHUNK 4096                // j-elements per block
#define TPB   256                 // 8 wave32s
#define PAIRS_PER_THREAD (CHUNK / 2 / TPB)   // 8

typedef unsigned int       u32;
typedef unsigned long long u64;
typedef u32   u32x4 __attribute__((ext_vector_type(4)));
typedef int   i32x4 __attribute__((ext_vector_type(4)));
typedef int   i32x8 __attribute__((ext_vector_type(8)));
typedef float f32x2 __attribute__((ext_vector_type(2)));
typedef float f32x4 __attribute__((ext_vector_type(4)));

__global__ void FreqProj_zero_kernel(float* out) {
    out[threadIdx.x] = 0.0f;
}

__launch_bounds__(TPB)
__global__ void FreqProj_main_kernel(const float* __restrict__ f,
                                     const float* __restrict__ x,
                                     const float* __restrict__ k,
                                     float* __restrict__ out)
{
    __shared__ float sh_k[2 * CHUNK];   // row 0: k0 chunk, row 1: k1 chunk
    __shared__ float sh_red[16];

    const int b  = blockIdx.y;
    const int j0 = blockIdx.x * CHUNK;
    const int t  = threadIdx.x;

    // ---- TDM: DMA k[0][j0 : j0+CHUNK) and k[1][j0 : j0+CHUNK) into LDS ----
    // 2-D tile: tile_dim0 = CHUNK, tile_dim1 = 2 rows, row stride = F2 elems.
    // Issued by wave 0 only (TDM is wave-level; branching keeps other waves out).
    if (t < 32) {
        const u64 gaddr = (u64)(uintptr_t)(k + j0);          // byte addr of tile start
        const u32 ldsa  = (u32)(uintptr_t)(&sh_k[0]);        // low 32 bits = LDS byte addr

        u32x4 g0;
        g0.x = 1u;                                           // count=1 valid descriptor
        g0.y = ldsa;                                         // lds_addr
        g0.z = (u32)(gaddr & 0xFFFFFFFFu);                   // global_addr[31:0]
        g0.w = (u32)((gaddr >> 32) & 0x01FFFFFFu) | (2u << 30); // addr[56:32] | type=2

        i32x8 g1;
        g1[0] = (int)(2u << 16);                             // data_size=4B, wg_mask=0
        g1[1] = (int)(((u32)F2 & 0xFFFFu) << 16);            // tensor_dim0[15:0] in [63:48]
        g1[2] = (int)((2u << 16) | ((u32)F2 >> 16));         // tensor_dim1=2 | tensor_dim0[31:16]
        g1[3] = (int)((u32)CHUNK << 16);                     // tile_dim0=CHUNK | tensor_dim1[31:16]=0
        g1[4] = (int)2u;                                     // tile_dim2=0 | tile_dim1=2
        g1[5] = (int)(u32)F2;                                // tensor_dim0_stride[31:0] = F2
        g1[6] = 0;                                           // dim1_stride[15:0] | dim0_stride[47:32]
        g1[7] = 0;

        i32x4 gz4 = {0, 0, 0, 0};                            // groups 2/3 unused (2-D tensor)
        i32x8 gz8 = {0, 0, 0, 0, 0, 0, 0, 0};                // extra group (clang-23 6-arg form)

        __builtin_amdgcn_tensor_load_to_lds(g0, g1, gz4, gz4, gz8, 0);
        __builtin_amdgcn_s_wait_tensorcnt(0);
    }
    __syncthreads();

    // ---- streaming reduction over this block's j-chunk ----
    const float x0   = x[2 * b];
    const float x1   = x[2 * b + 1];
    const float invF = 1.0f / (float)FGRID;   // phase/(2pi) = (x.k)/F : revolutions for v_sin/v_cos

    const f32x4* __restrict__ F0 =
        (const f32x4*)(f + (((size_t)b * 2 + 0) * (size_t)F2 + (size_t)j0) * 2);
    const f32x4* __restrict__ F1 =
        (const f32x4*)(f + (((size_t)b * 2 + 1) * (size_t)F2 + (size_t)j0) * 2);
    const f32x2* __restrict__ K0 = (const f32x2*)(sh_k);
    const f32x2* __restrict__ K1 = (const f32x2*)(sh_k + CHUNK);

    float acc0 = 0.0f, acc1 = 0.0f;
#pragma unroll
    for (int it = 0; it < PAIRS_PER_THREAD; ++it) {
        const int p = t + it * TPB;                  // pair index: covers j = j0+2p, j0+2p+1
        // read-once 512MB stream: non-temporal B128 loads (keep k hot in L2)
        f32x4 a = __builtin_nontemporal_load(&F0[p]);   // {re0, im0, re1, im1} channel 0
        f32x4 c = __builtin_nontemporal_load(&F1[p]);   // channel 1
        f32x2 q0 = K0[p];                               // k0[j], k0[j+1] from LDS
        f32x2 q1 = K1[p];                               // k1[j], k1[j+1] from LDS

        // r in revolutions -> hardware v_cos_f32 / v_sin_f32, no range reduction
        float r0 = __builtin_fmaf(x1, q1.x, x0 * q0.x) * invF;
        float r1 = __builtin_fmaf(x1, q1.y, x0 * q0.y) * invF;
        float cr0 = __builtin_amdgcn_cosf(r0);
        float si0 = __builtin_amdgcn_sinf(r0);
        float cr1 = __builtin_amdgcn_cosf(r1);
        float si1 = __builtin_amdgcn_sinf(r1);

        acc0 = __builtin_fmaf(a.x, cr0, __builtin_fmaf(-a.y, si0, acc0));
        acc0 = __builtin_fmaf(a.z, cr1, __builtin_fmaf(-a.w, si1, acc0));
        acc1 = __builtin_fmaf(c.x, cr0, __builtin_fmaf(-c.y, si0, acc1));
        acc1 = __builtin_fmaf(c.z, cr1, __builtin_fmaf(-c.w, si1, acc1));
    }

    // ---- wave32 tree reduction, then cross-wave via LDS ----
#pragma unroll
    for (int off = 16; off > 0; off >>= 1) {
        acc0 += __shfl_xor(acc0, off, 32);
        acc1 += __shfl_xor(acc1, off, 32);
    }
    const int wave = t >> 5;
    if ((t & 31) == 0) {
        sh_red[wave]     = acc0;
        sh_red[8 + wave] = acc1;
    }
    __syncthreads();
    if (t == 0) {
        float s0 = 0.0f, s1 = 0.0f;
#pragma unroll
        for (int w = 0; w < 8; ++w) { s0 += sh_red[w]; s1 += sh_red[8 + w]; }
        const float scale = 1.0f / (float)F2;   // mean over frequency axis
        atomicAdd(&out[b * 2 + 0], s0 * scale);
        atomicAdd(&out[b * 2 + 1], s1 * scale);
    }
}

extern "C" void kernel_launch(void* const* d_in, const int* in_sizes, int n_in,
                              void* d_out, int out_size, void* d_ws, size_t ws_size,
                              hipStream_t stream) {
    (void)in_sizes; (void)n_in; (void)d_ws; (void)ws_size; (void)out_size;
    // setup_inputs order: f [B,2,F^2,2] f32 | x [B,2] f32 | t [B] f32 (unused) | k [2,F^2] f32
    const float* f = (const float*)d_in[0];
    const float* x = (const float*)d_in[1];
    const float* k = (const float*)d_in[3];
    float* out = (float*)d_out;   // [B,2] = 64 floats

    FreqProj_zero_kernel<<<1, BATCH * 2, 0, stream>>>(out);

    dim3 grid(F2 / CHUNK, BATCH, 1);   // 256 x 32 = 8192 blocks
    FreqProj_main_kernel<<<grid, TPB, 0, stream>>>(f, x, k, out);
}